// CTProjector3DModule_36369783063164
// MI455X (gfx1250) — compile-verified
//
#include <hip/hip_runtime.h>
#include <math.h>
#include <stdint.h>

// 3D cone-beam Siddon forward projector for MI455X (gfx1250, wave32).
// tvals (203 MB, streamed once) is tiled into LDS with the Tensor Data Mover
// (TENSOR_LOAD_TO_LDS, double-buffered, TENSORcnt-synchronized); the 16 MB
// volume stays L2-resident and is gathered with z-coalesced loads.

#define CT_NX 128
#define CT_NY 128
#define CT_NZ 128
#define BLOCK 256
#define CH    31   // tile width in dwords; gcd(31,64)=1 -> conflict-free LDS rows; 62KB LDS total

typedef __attribute__((ext_vector_type(4))) unsigned int v4u;
typedef __attribute__((ext_vector_type(8))) int          v8i;
typedef __attribute__((ext_vector_type(4))) int          v4i;

#if defined(__has_builtin)
#if __has_builtin(__builtin_amdgcn_tensor_load_to_lds) && __has_builtin(__builtin_amdgcn_s_wait_tensorcnt)
#define HAVE_TDM 1
#endif
#endif
#ifndef HAVE_TDM
#define HAVE_TDM 0
#endif

#if __has_include(<hip/amd_detail/amd_gfx1250_TDM.h>)
#define TDM_6ARG 1
#else
#define TDM_6ARG 0
#endif

// Low 32 bits of a flat shared-memory pointer = byte offset in the wave's LDS.
__device__ __forceinline__ unsigned lds_lo32(const void* p) {
    return (unsigned)(uintptr_t)p;
}

#if HAVE_TDM
// Build the D# (ISA 8.3/8.4) for a 2D dword tile and issue TENSOR_LOAD_TO_LDS.
// tensor_dim == tile_dim so every element is in-bounds by construction.
__device__ __forceinline__ void tdm_load_tile(unsigned lds_byte,
                                              unsigned long long gaddr,
                                              unsigned tile_w, unsigned tile_h,
                                              unsigned long long stride_elems)
{
    v4u g0;
    g0.x = 1u;                                                   // count=1 (valid user D#)
    g0.y = lds_byte;                                             // lds_addr [63:32]
    g0.z = (unsigned)(gaddr & 0xffffffffull);                    // global_addr [95:64]
    g0.w = (unsigned)((gaddr >> 32) & 0x01ffffffull) | (2u << 30); // addr[56:32] + type=2

    v8i g1;
    g1[0] = (int)(2u << 16);                                     // data_size=4B; mask=0; no pad/iterate
    g1[1] = (int)((tile_w & 0xffffu) << 16);                     // tensor_dim0[15:0] @ bits 63:48
    g1[2] = (int)(((tile_w >> 16) & 0xffffu) | ((tile_h & 0xffffu) << 16)); // td0 hi | td1 lo
    g1[3] = (int)(((tile_h >> 16) & 0xffffu) | ((tile_w & 0xffffu) << 16)); // td1 hi | tile_dim0
    g1[4] = (int)(tile_h & 0xffffu);                             // tile_dim1 (tile_dim2=0)
    g1[5] = (int)(unsigned)(stride_elems & 0xffffffffull);       // tensor_dim0_stride lo
    g1[6] = (int)(unsigned)((stride_elems >> 32) & 0xffffull);   // stride hi (dim1_stride=0)
    g1[7] = 0;

    v4i z4 = (v4i){0, 0, 0, 0};
#if TDM_6ARG
    v8i z8 = (v8i){0, 0, 0, 0, 0, 0, 0, 0};
    __builtin_amdgcn_tensor_load_to_lds(g0, g1, z4, z4, z8, 0);
#else
    __builtin_amdgcn_tensor_load_to_lds(g0, g1, z4, z4, 0);
#endif
}
#endif  // HAVE_TDM

template <int BATCH>
__device__ __forceinline__ void seg_accum(float t0, float t1,
                                          float ax, float ay, float az,
                                          float dx, float dy, float dz,
                                          float raylen,
                                          const float* __restrict__ vol,
                                          float (&acc)[BATCH])
{
    if (t1 > t0) {
        const float tm = 0.5f * (t0 + t1);
        const float px = ax + tm * dx;
        const float py = ay + tm * dy;
        const float pz = az + tm * dz;
        const int ix = (int)floorf(px);
        const int iy = (int)floorf(py);
        const int iz = (int)floorf(pz);
        if ((unsigned)ix < (unsigned)CT_NX &&
            (unsigned)iy < (unsigned)CT_NY &&
            (unsigned)iz < (unsigned)CT_NZ) {
            const int flat = (ix << 14) + (iy << 7) + iz;  // 128^3
            const float w = (t1 - t0) * raylen;
#pragma unroll
            for (int q = 0; q < BATCH; ++q)
                acc[q] = fmaf(w, vol[(size_t)q * (CT_NX * CT_NY * CT_NZ) + flat], acc[q]);
        }
    }
}

template <int BATCH>
__global__ __launch_bounds__(BLOCK)
void ct_siddon_fwd(const float* __restrict__ vol,     // (BATCH,128,128,128)
                   const float* __restrict__ tvals,   // (R,S) sorted, +inf tail
                   const float* __restrict__ Mm,      // (3,3)
                   const float* __restrict__ bvec,    // (3,)
                   const float* __restrict__ src,     // (R,3)
                   const float* __restrict__ dst,     // (R,3)
                   float* __restrict__ out,           // (BATCH,R)
                   int R, int S)
{
#if HAVE_TDM
    __shared__ float tile[2][BLOCK * CH];
#endif
    const int tid     = threadIdx.x;
    const int rayBase = blockIdx.x * BLOCK;
    const int r       = rayBase + tid;
    const int rr      = (r < R) ? r : (R - 1);

    // Uniform geometry (s_load).
    const float m00 = Mm[0], m01 = Mm[1], m02 = Mm[2];
    const float m10 = Mm[3], m11 = Mm[4], m12 = Mm[5];
    const float m20 = Mm[6], m21 = Mm[7], m22 = Mm[8];
    const float b0 = bvec[0], b1 = bvec[1], b2 = bvec[2];

    const float sx = src[3 * rr + 0], sy = src[3 * rr + 1], sz = src[3 * rr + 2];
    const float ex = dst[3 * rr + 0] - sx;
    const float ey = dst[3 * rr + 1] - sy;
    const float ez = dst[3 * rr + 2] - sz;

    const float ax = m00 * sx + m01 * sy + m02 * sz + b0;
    const float ay = m10 * sx + m11 * sy + m12 * sz + b1;
    const float az = m20 * sx + m21 * sy + m22 * sz + b2;
    const float dx = m00 * ex + m01 * ey + m02 * ez;
    const float dy = m10 * ex + m11 * ey + m12 * ez;
    const float dz = m20 * ex + m21 * ey + m22 * ez;
    const float raylen = sqrtf(ex * ex + ey * ey + ez * ez);

    const float* __restrict__ row = tvals + (size_t)rr * (size_t)S;

    float acc[BATCH];
#pragma unroll
    for (int q = 0; q < BATCH; ++q) acc[q] = 0.0f;

    float t0  = __builtin_nontemporal_load(row);   // row[0]
    bool done = !(t0 < 3.0e38f);                   // +inf -> no valid segments
    int  sTail = 1;                                // first column not yet processed

#if HAVE_TDM
    const bool fullBlock = (rayBase + BLOCK) <= R;
    if (fullBlock) {
        const int nfull = (S - 1) / CH;            // TDM-staged chunks of CH columns
        sTail = 1 + nfull * CH;
        const unsigned long long tbase = (unsigned long long)(uintptr_t)tvals;

        if (tid < 32 && nfull > 0) {
            tdm_load_tile(lds_lo32(&tile[0][0]),
                          tbase + ((unsigned long long)rayBase * S + 1ull) * 4ull,
                          CH, BLOCK, (unsigned long long)S);
        }
        for (int c = 0; c < nfull; ++c) {
            if (tid < 32) {
                if (c + 1 < nfull) {
                    tdm_load_tile(lds_lo32(&tile[(c + 1) & 1][0]),
                                  tbase + ((unsigned long long)rayBase * S + 1ull +
                                           (unsigned long long)(c + 1) * CH) * 4ull,
                                  CH, BLOCK, (unsigned long long)S);
                    __builtin_amdgcn_s_wait_tensorcnt(1);  // chunk c landed (in-order)
                } else {
                    __builtin_amdgcn_s_wait_tensorcnt(0);
                }
            }
            __syncthreads();                        // publish buffer c&1 to all waves
            if (!done) {
                const float* __restrict__ lrow = &tile[c & 1][tid * CH];
                for (int j = 0; j < CH; ++j) {
                    const float t1 = lrow[j];
                    if (!(t1 < 3.0e38f)) { done = true; break; }
                    seg_accum<BATCH>(t0, t1, ax, ay, az, dx, dy, dz, raylen, vol, acc);
                    t0 = t1;
                }
            }
            // Barrier doubles as block-wide early-exit vote (rays in a block
            // share one detector column -> near-identical segment counts).
            if (__syncthreads_and(done ? 1 : 0)) { done = true; break; }
        }
    }
#endif  // HAVE_TDM

    // Tail columns (and full fallback path when TDM unavailable / partial block).
    if (!done && r < R) {
        for (int s = sTail; s < S; ++s) {
            if ((s & 31) == 1) __builtin_prefetch(row + s + 64, 0, 0);
            const float t1 = __builtin_nontemporal_load(row + s);
            if (!(t1 < 3.0e38f)) break;
            seg_accum<BATCH>(t0, t1, ax, ay, az, dx, dy, dz, raylen, vol, acc);
            t0 = t1;
        }
    }

    if (r < R) {
#pragma unroll
        for (int q = 0; q < BATCH; ++q)
            out[(size_t)q * (size_t)R + r] = acc[q];
    }
}

extern "C" void kernel_launch(void* const* d_in, const int* in_sizes, int n_in,
                              void* d_out, int out_size, void* d_ws, size_t ws_size,
                              hipStream_t stream)
{
    // setup_inputs() order: volume, tvals, M, b, src, dst
    const float* vol   = (const float*)d_in[0];
    const float* tvals = (const float*)d_in[1];
    const float* Mm    = (const float*)d_in[2];
    const float* bb    = (const float*)d_in[3];
    const float* src   = (const float*)d_in[4];
    const float* dst   = (const float*)d_in[5];
    float* out = (float*)d_out;

    const int R = in_sizes[4] / 3;                        // 131072 rays
    const int S = in_sizes[1] / R;                        // 387 t-values per ray
    const int B = in_sizes[0] / (CT_NX * CT_NY * CT_NZ);  // batch (2)

    dim3 block(BLOCK);
    dim3 grid((unsigned)((R + BLOCK - 1) / BLOCK));

    if (B == 2) {
        ct_siddon_fwd<2><<<grid, block, 0, stream>>>(vol, tvals, Mm, bb, src, dst, out, R, S);
    } else {
        ct_siddon_fwd<1><<<grid, block, 0, stream>>>(vol, tvals, Mm, bb, src, dst, out, R, S);
    }
}